// SSIMLoss_21775484191476
// MI455X (gfx1250) — compile-verified
//
#include <hip/hip_runtime.h>
#include <math.h>

// ---------------- problem constants ----------------
#define WINDOW   11
#define HALO     5
#define IMG_H    256
#define IMG_W    256
#define T_FRAMES 16
#define B_SAMPLES 8
#define PER_SAMPLE (T_FRAMES * IMG_H * IMG_W)   // 1,048,576
#define NCHUNK   64
#define CHUNK    (PER_SAMPLE / NCHUNK)          // 16,384

// ---------------- tile geometry --------------------
#define TW 64                 // output tile width
#define TH 32                 // output tile height
#define LW (TW + 2 * HALO)    // 74 input cols
#define LH (TH + 2 * HALO)    // 42 input rows
#define LSTRIDE 76            // padded LDS row stride (covers K-pad cols 74,75)

typedef __attribute__((ext_vector_type(2))) float v2f;
typedef __attribute__((ext_vector_type(8))) float v8f;

// Normalized 11-tap Gaussian, sigma = 1.5 (sums to 1.0)
__device__ __constant__ float GW[WINDOW] = {
    0.00102838f, 0.00759877f, 0.03600077f, 0.10936069f, 0.21300553f,
    0.26601172f, 0.21300553f, 0.10936069f, 0.03600077f, 0.00759877f,
    0.00102838f};

// Banded horizontal-conv matrix G[j][n] = g[j-n] for 0<=j-n<=10, j<26, else 0.
__device__ __forceinline__ float gband(int j, int n) {
    int d = j - n;
    return (j < 26 && d >= 0 && d <= 10) ? GW[d] : 0.0f;
}

// ============ Kernel 1: chunked per-sample stats ============
__global__ __launch_bounds__(256) void stats_partial(
    const float* __restrict__ pred, const float* __restrict__ targ,
    float* __restrict__ part) {
    const int chunk = blockIdx.x, b = blockIdx.y, tid = threadIdx.x;
    const size_t base = (size_t)b * PER_SAMPLE + (size_t)chunk * CHUNK;
    float mnp = 3.4e38f, mxp = -3.4e38f, mnt = 3.4e38f, mxt = -3.4e38f,
          st = 0.0f;
    for (int i = tid; i < CHUNK; i += 256) {
        float p = pred[base + i], t = targ[base + i];
        mnp = fminf(mnp, p); mxp = fmaxf(mxp, p);
        mnt = fminf(mnt, t); mxt = fmaxf(mxt, t);
        st += t;
    }
    __shared__ float sm[5][256];
    sm[0][tid] = mnp; sm[1][tid] = mxp; sm[2][tid] = mnt;
    sm[3][tid] = mxt; sm[4][tid] = st;
    __syncthreads();
    for (int s = 128; s > 0; s >>= 1) {
        if (tid < s) {
            sm[0][tid] = fminf(sm[0][tid], sm[0][tid + s]);
            sm[1][tid] = fmaxf(sm[1][tid], sm[1][tid + s]);
            sm[2][tid] = fminf(sm[2][tid], sm[2][tid + s]);
            sm[3][tid] = fmaxf(sm[3][tid], sm[3][tid + s]);
            sm[4][tid] += sm[4][tid + s];
        }
        __syncthreads();
    }
    if (tid == 0) {
        float* o = part + (size_t)(b * NCHUNK + chunk) * 5;
        o[0] = sm[0][0]; o[1] = sm[1][0]; o[2] = sm[2][0];
        o[3] = sm[3][0]; o[4] = sm[4][0];
    }
}

// ============ Kernel 2: finalize stats + zero accumulators ============
__global__ __launch_bounds__(64) void stats_final(
    const float* __restrict__ part, float* __restrict__ stats,
    float* __restrict__ ssim_sum) {
    const int b = blockIdx.x, tid = threadIdx.x;  // 64 threads
    __shared__ float sm[5][64];
    const float* p = part + (size_t)(b * NCHUNK + tid) * 5;
    sm[0][tid] = p[0]; sm[1][tid] = p[1]; sm[2][tid] = p[2];
    sm[3][tid] = p[3]; sm[4][tid] = p[4];
    __syncthreads();
    for (int s = 32; s > 0; s >>= 1) {
        if (tid < s) {
            sm[0][tid] = fminf(sm[0][tid], sm[0][tid + s]);
            sm[1][tid] = fmaxf(sm[1][tid], sm[1][tid + s]);
            sm[2][tid] = fminf(sm[2][tid], sm[2][tid + s]);
            sm[3][tid] = fmaxf(sm[3][tid], sm[3][tid + s]);
            sm[4][tid] += sm[4][tid + s];
        }
        __syncthreads();
    }
    if (tid == 0) {
        const float eps = 1e-6f;
        float invp = 1.0f / fmaxf(sm[1][0] - sm[0][0], eps);
        float invt = 1.0f / fmaxf(sm[3][0] - sm[2][0], eps);
        stats[b * 6 + 0] = sm[0][0];  // minP
        stats[b * 6 + 1] = invp;
        stats[b * 6 + 2] = sm[2][0];  // minT
        stats[b * 6 + 3] = invt;
        stats[b * 6 + 4] = (sm[4][0] != 0.0f) ? 1.0f : 0.0f;  // valid
        stats[b * 6 + 5] = 0.0f;
        ssim_sum[b] = 0.0f;  // zero accumulator (deterministic per launch)
    }
}

// ============ Kernel 3: fused normalize + 5x Gaussian blur + SSIM ============
// Block = 256 threads (8 waves). Output tile 32x64 -> 8 subtiles of 16x16,
// one per wave. Vertical 11-tap blur in VALU -> LDS; horizontal 11-tap blur
// as banded GEMM via V_WMMA_F32_16X16X4_F32 (K = 28, 7 chained WMMAs).
// Field loop fully unrolled: 5 fields x 7 = 35 WMMAs, no runtime selects.
__global__ __launch_bounds__(256) void ssim_tile(
    const float* __restrict__ pred, const float* __restrict__ targ,
    const float* __restrict__ stats, float* __restrict__ ssim_sum) {
    const int tid = threadIdx.x;
    const int tile_c0 = blockIdx.x * TW;
    const int tile_r0 = blockIdx.y * TH;
    const int n = blockIdx.z;   // image index in [0,128)
    const int b = n >> 4;       // sample index
    const size_t img = (size_t)n * (IMG_H * IMG_W);

    const float mnp = stats[b * 6 + 0], invp = stats[b * 6 + 1];
    const float mnt = stats[b * 6 + 2], invt = stats[b * 6 + 3];

    __shared__ float s_t[LH * LSTRIDE];   // normalized target halo tile
    __shared__ float s_p[LH * LSTRIDE];   // normalized pred halo tile
    __shared__ float s_v[TH * LSTRIDE];   // vertical-blur scratch (one field)
    __shared__ float s_red[8];

    // Load + normalize halo tile (zero padding outside image)
    for (int idx = tid; idx < LH * LW; idx += 256) {
        int i = idx / LW, j = idx % LW;
        int gr = tile_r0 - HALO + i, gc = tile_c0 - HALO + j;
        float tv = 0.0f, pv = 0.0f;
        if (gr >= 0 && gr < IMG_H && gc >= 0 && gc < IMG_W) {
            size_t a = img + (size_t)gr * IMG_W + gc;
            tv = (targ[a] - mnt) * invt;
            pv = (pred[a] - mnp) * invp;
        }
        s_t[i * LSTRIDE + j] = tv;
        s_p[i * LSTRIDE + j] = pv;
    }
    __syncthreads();

    // Per-wave WMMA subtile coordinates
    const int lane = tid & 31;
    const int wave = tid >> 5;
    const int rowbase = (wave >> 2) * 16;  // 0 or 16
    const int colbase = (wave & 3) * 16;   // 0,16,32,48
    const int m = lane & 15;               // A row / B col for this lane
    const int kk = (lane >> 4) * 2;        // K sub-offset per lane half

    // B-matrix (banded Gaussian) values are field-independent: precompute.
    float bv0[7], bv1[7];
#pragma unroll
    for (int c = 0; c < 7; ++c) {
        bv0[c] = gband(c * 4 + kk, m);
        bv1[c] = gband(c * 4 + kk + 1, m);
    }

    const float* arow = s_v + (rowbase + m) * LSTRIDE + colbase;
    v8f acc[5];

#pragma unroll
    for (int f = 0; f < 5; ++f) {
        // ---- vertical 11-tap blur of field f into s_v (cols>=LW zeroed) ----
        // f is compile-time after unroll -> the field select folds away.
        for (int idx = tid; idx < TH * LSTRIDE; idx += 256) {
            int i = idx / LSTRIDE, j = idx - i * LSTRIDE;
            float vsum = 0.0f;
            if (j < LW) {
#pragma unroll
                for (int k = 0; k < WINDOW; ++k) {
                    float a = s_t[(i + k) * LSTRIDE + j];
                    float c = s_p[(i + k) * LSTRIDE + j];
                    float val = (f == 0) ? a
                              : (f == 1) ? c
                              : (f == 2) ? a * a
                              : (f == 3) ? c * c
                              :            a * c;
                    vsum = fmaf(GW[k], val, vsum);
                }
            }
            s_v[idx] = vsum;
        }
        __syncthreads();

        // ---- horizontal blur: D(16x16) = A(16x28) x G(28x16) via WMMA ----
        v8f d = {};
#pragma unroll
        for (int c = 0; c < 7; ++c) {
            int ka = c * 4 + kk;
            v2f va = {arow[ka], arow[ka + 1]};
            v2f vb = {bv0[c], bv1[c]};
            d = __builtin_amdgcn_wmma_f32_16x16x4_f32(
                false, va, false, vb, (short)0, d, false, false);
        }
        acc[f] = d;
        __syncthreads();  // protect s_v before next field overwrites it
    }

    // ---- per-pixel SSIM on this wave's 16x16 tile (8 elems per lane) ----
    // Single divide: lum*con*str = (num1*num2*num3) / (den1*den2*den3)
    const float C1 = 1.0e-4f;   // (0.01)^2
    const float C2 = 9.0e-4f;   // (0.03)^2
    const float C3 = 4.5e-4f;   // C2/2
    const float eps = 1e-6f;
    float sum = 0.0f;
#pragma unroll
    for (int e = 0; e < 8; ++e) {
        float m1 = acc[0][e], m2 = acc[1][e];
        float m1sq = m1 * m1, m2sq = m2 * m2, m12 = m1 * m2;
        float s1 = fmaxf(acc[2][e] - m1sq, eps);
        float s2 = fmaxf(acc[3][e] - m2sq, eps);
        float s12 = acc[4][e] - m12;
        float root = sqrtf(s1 * s2);
        float num = (2.0f * m12 + C1) * (2.0f * root + C2) * (s12 + C3);
        float den = (m1sq + m2sq + C1) * (s1 + s2 + C2) * (root + C3);
        sum += num / den;
    }

    // wave32 butterfly reduce, then 8 wave partials through LDS
#pragma unroll
    for (int off = 16; off > 0; off >>= 1)
        sum += __shfl_xor(sum, off, 32);
    if (lane == 0) s_red[wave] = sum;
    __syncthreads();
    if (tid == 0) {
        float bs = 0.0f;
#pragma unroll
        for (int w = 0; w < 8; ++w) bs += s_red[w];
        atomicAdd(&ssim_sum[b], bs);
    }
}

// ============ Kernel 4: final weighted loss ============
__global__ void final_loss(const float* __restrict__ stats,
                           const float* __restrict__ ssim_sum,
                           float* __restrict__ out) {
    if (threadIdx.x == 0 && blockIdx.x == 0) {
        float num = 0.0f, nvalid = 0.0f;
        const float inv_px = 1.0f / (float)(T_FRAMES * IMG_H * IMG_W);
        for (int b = 0; b < B_SAMPLES; ++b) {
            float valid = stats[b * 6 + 4];
            float sps = ssim_sum[b] * inv_px;
            num += (1.0f - sps) * valid;
            nvalid += valid;
        }
        out[0] = num / fmaxf(nvalid, 1.0f);
    }
}

// ============ launch ============
extern "C" void kernel_launch(void* const* d_in, const int* in_sizes, int n_in,
                              void* d_out, int out_size, void* d_ws,
                              size_t ws_size, hipStream_t stream) {
    const float* pred = (const float*)d_in[0];
    const float* targ = (const float*)d_in[1];
    float* ws = (float*)d_ws;
    float* part  = ws;                // 8*64*5 = 2560 floats
    float* stats = ws + 2560;         // 8*6    = 48 floats
    float* ssum  = ws + 2560 + 48;    // 8 floats
    float* out = (float*)d_out;

    dim3 g1(NCHUNK, B_SAMPLES);
    stats_partial<<<g1, 256, 0, stream>>>(pred, targ, part);
    stats_final<<<B_SAMPLES, 64, 0, stream>>>(part, stats, ssum);
    dim3 g2(IMG_W / TW, IMG_H / TH, B_SAMPLES * T_FRAMES);  // (4, 8, 128)
    ssim_tile<<<g2, 256, 0, stream>>>(pred, targ, stats, ssum);
    final_loss<<<1, 32, 0, stream>>>(stats, ssum, out);
}